// GraphPool_58102317580658
// MI455X (gfx1250) — compile-verified
//
#include <hip/hip_runtime.h>
#include <hip/hip_bf16.h>

// GraphPool (top-k node pooling), MI455X / gfx1250, wave32.
// x:(B=16,N=4096,D=1024) f32, w:(D) f32, b scalar.
// scores = sigmoid(x.w + b); top-k (k=N/2) descending with index tie-break;
// out[b,j,:] = x[b,idx,:] * scores[b,idx].
//
// HBM-bound: ~512 MB total traffic -> ~22us at 23.3 TB/s. The GEMV runs on
// V_WMMA_F32_16X16X4_F32 (16 row-dots per wave-instruction, f32 exact).

typedef __attribute__((ext_vector_type(2))) float v2f;
typedef __attribute__((ext_vector_type(8))) float v8f;

#define GP_B 16
#define GP_N 4096
#define GP_D 1024
#define GP_K 2048   // N * POOLING_RATIO

// ---------------------------------------------------------------------------
// Kernel 1: scores[b*N + n] = sigmoid(dot(x[row,:], w) + bias)
// One wave handles 16 consecutive rows. Per step: A = 16x4 f32 tile of x,
// B = 4x16 with w[k..k+3] replicated into every column, accumulate f32 16x16.
// All 16 columns of D are identical = the row dots.
//
// ISA 16x4 f32 A layout: lane(0..15) row M=lane, VGPR0/1 = K 0/1;
// lanes 16..31 same rows, VGPR0/1 = K 2/3.
// ISA 4x16 f32 B layout: lanes 0..15 col N=lane, VGPR0/1 = K 0/1;
// lanes 16..31 cols, K 2/3. Replicated w => value depends only on lane>>4.
// C/D f32 layout: lanes 0..15 N=lane, VGPR v = M=v; lanes 16..31 VGPR v = M=v+8.
// ---------------------------------------------------------------------------
__global__ __launch_bounds__(256)
void gp_scores_wmma(const float* __restrict__ x,
                    const float* __restrict__ w,
                    const float* __restrict__ bias,
                    float* __restrict__ scores)
{
    const unsigned lane = threadIdx.x & 31u;
    const unsigned wave = threadIdx.x >> 5;          // 0..7
    const unsigned gw   = blockIdx.x * 8u + wave;    // global wave id
    const unsigned row0 = gw * 16u;                  // first of 16 rows (flat over B*N)
    if (row0 >= GP_B * GP_N) return;

    const unsigned m    = lane & 15u;                // row within tile
    const unsigned half = lane >> 4;                 // 0: K0/1, 1: K2/3

    const float* arow = x + (size_t)(row0 + m) * GP_D + (half << 1);
    const float* wrow = w + (half << 1);

    v8f acc = {};
#pragma unroll 4
    for (int k = 0; k < GP_D; k += 4) {
        v2f a = *(const v2f*)(arow + k);
        v2f bb = *(const v2f*)(wrow + k);
        // D = A(16x4) * B(4x16) + C, f32 exact accumulation
        acc = __builtin_amdgcn_wmma_f32_16x16x4_f32(
            /*neg_a=*/false, a, /*neg_b=*/false, bb,
            /*c_mod=*/(short)0, acc, /*reuse_a=*/false, /*reuse_b=*/false);
    }

    const float bv = bias[0];
    // Column 0 lives in lane 0 (rows 0..7) and lane 16 (rows 8..15).
    if ((lane & 15u) == 0u) {
        const unsigned mbase = (half == 0) ? 0u : 8u;
#pragma unroll
        for (int v = 0; v < 8; ++v) {
            float d = acc[v] + bv;
            float s = 1.0f / (1.0f + __expf(-d));
            scores[row0 + mbase + (unsigned)v] = s;
        }
    }
}

// ---------------------------------------------------------------------------
// Kernel 2: per-batch descending bitonic sort of (score, index) composite
// keys in LDS; emit top-k indices. key = ord(score)<<32 | (~idx) so that
// descending u64 order == (score desc, index asc), matching jax.lax.top_k.
// ---------------------------------------------------------------------------
__global__ __launch_bounds__(1024)
void gp_topk_bitonic(const float* __restrict__ scores,
                     int* __restrict__ top_idx)
{
    __shared__ unsigned long long keys[GP_N];   // 32 KB of the 320 KB LDS
    const int bat = blockIdx.x;
    const float* s = scores + (size_t)bat * GP_N;

    for (unsigned i = threadIdx.x; i < GP_N; i += blockDim.x) {
        unsigned u = __float_as_uint(s[i]);
        u = (u & 0x80000000u) ? ~u : (u | 0x80000000u);   // order-preserving map
        keys[i] = ((unsigned long long)u << 32) |
                  (unsigned long long)(0xFFFFFFFFu - i);
    }
    __syncthreads();

    for (unsigned size = 2; size <= GP_N; size <<= 1) {
        for (unsigned stride = size >> 1; stride > 0; stride >>= 1) {
            for (unsigned p = threadIdx.x; p < (GP_N / 2); p += blockDim.x) {
                unsigned i = ((p & ~(stride - 1u)) << 1) | (p & (stride - 1u));
                unsigned j = i + stride;
                bool desc = ((i & size) == 0u);
                unsigned long long a = keys[i];
                unsigned long long c = keys[j];
                if ((a < c) == desc) { keys[i] = c; keys[j] = a; }
            }
            __syncthreads();
        }
    }

    for (unsigned t = threadIdx.x; t < GP_K; t += blockDim.x) {
        unsigned idx = 0xFFFFFFFFu - (unsigned)(keys[t] & 0xFFFFFFFFull);
        top_idx[bat * GP_K + t] = (int)idx;
    }
}

// ---------------------------------------------------------------------------
// Kernel 3: out[b,j,:] = x[b, top_idx[b,j], :] * scores[b, top_idx[b,j]]
// One block per output row; float4 coalesced streaming.
// ---------------------------------------------------------------------------
__global__ __launch_bounds__(256)
void gp_gather_scale(const float* __restrict__ x,
                     const float* __restrict__ scores,
                     const int* __restrict__ top_idx,
                     float* __restrict__ out)
{
    const unsigned row = blockIdx.x;           // 0 .. B*K-1
    const unsigned bat = row / GP_K;
    const unsigned j   = row - bat * GP_K;
    const int idx = top_idx[bat * GP_K + j];
    const float sc = scores[(size_t)bat * GP_N + (unsigned)idx];

    const float4* src = (const float4*)(x + ((size_t)bat * GP_N + (unsigned)idx) * GP_D);
    float4* dst = (float4*)(out + (size_t)row * GP_D);

    const unsigned t = threadIdx.x;            // 256 threads * float4 = 1024 floats
    float4 v = src[t];
    v.x *= sc; v.y *= sc; v.z *= sc; v.w *= sc;
    dst[t] = v;
}

// ---------------------------------------------------------------------------
extern "C" void kernel_launch(void* const* d_in, const int* in_sizes, int n_in,
                              void* d_out, int out_size, void* d_ws, size_t ws_size,
                              hipStream_t stream)
{
    const float* x  = (const float*)d_in[0];
    const float* w  = (const float*)d_in[1];
    const float* b  = (const float*)d_in[2];
    float* out      = (float*)d_out;

    // workspace: scores (B*N f32 = 256 KB) then top_idx (B*K i32 = 128 KB)
    float* scores = (float*)d_ws;
    int*   topidx = (int*)((char*)d_ws + (size_t)GP_B * GP_N * sizeof(float));

    // 1) scores: (B*N)/16 waves, 8 waves per block -> 512 blocks
    const int waves  = (GP_B * GP_N) / 16;
    const int blocks = waves / 8;
    gp_scores_wmma<<<blocks, 256, 0, stream>>>(x, w, b, scores);

    // 2) per-batch top-k via LDS bitonic sort
    gp_topk_bitonic<<<GP_B, 1024, 0, stream>>>(scores, topidx);

    // 3) gather + scale: one block per output row
    gp_gather_scale<<<GP_B * GP_K, 256, 0, stream>>>(x, scores, topidx, out);
}